// MyModel_13657996001718
// MI455X (gfx1250) — compile-verified
//
#include <hip/hip_runtime.h>

typedef __attribute__((ext_vector_type(2))) float v2f;
typedef __attribute__((ext_vector_type(4))) float v4f;
typedef __attribute__((ext_vector_type(8))) float v8f;

#define T_STEPS 200
#define HID 4
#define OUTN 28
#define WP_ENTRIES (T_STEPS * 2 * 32)   // [t][kpair][col] float2 entries
#define SWAPX16 0x401f                  // ds_swizzle: xor=0x10, or=0, and=0x1f

// ---------- fast activations (gfx1250 has v_tanh_f32) ----------
__device__ __forceinline__ float fast_tanh(float x) {
#if __has_builtin(__builtin_amdgcn_tanhf)
  return __builtin_amdgcn_tanhf(x);
#else
  float e = __builtin_amdgcn_exp2f(-2.8853900817779268f * x);
  return (1.0f - e) * __builtin_amdgcn_rcpf(1.0f + e);
#endif
}
__device__ __forceinline__ float fast_sigmoid(float x) {
  return __builtin_fmaf(0.5f, fast_tanh(0.5f * x), 0.5f);
}
__device__ __forceinline__ v2f tanh2(v2f v) {
  v2f r; r.x = fast_tanh(v.x); r.y = fast_tanh(v.y); return r;
}
__device__ __forceinline__ v2f sig2(v2f v) {
  v2f r; r.x = fast_sigmoid(v.x); r.y = fast_sigmoid(v.y); return r;
}
__device__ __forceinline__ float swap16(float v) {
  int i = __builtin_bit_cast(int, v);
  i = __builtin_amdgcn_ds_swizzle(i, SWAPX16);
  return __builtin_bit_cast(float, i);
}

// ---------- one LSTM cell step, gate-pair packed (i,f,g,o order) ----------
// pw: 32 x v4f = {W[2p][2kq], W[2p+1][2kq], W[2p][2kq+1], W[2p+1][2kq+1]}, idx p*4+kq
//     where W[j][k] = k<4 ? Wih[j][k] : Whh[j][k-4]
// pbv: 4 x v4f = bias (bih+bhh) in natural gate order
// in : {x0,x1,x2,x3,h0,h1,h2,h3};  h,c: 2 x v2f each (pairs {0,1},{2,3})
__device__ __forceinline__ void cell_pk(const v4f* pw, const v4f* pbv,
                                        const float in[8], v2f h[2], v2f c[2]) {
  v2f g[8];
#pragma unroll
  for (int q = 0; q < 4; ++q) {
    v4f bq = pbv[q];
    g[2 * q].x = bq.x;     g[2 * q].y = bq.y;
    g[2 * q + 1].x = bq.z; g[2 * q + 1].y = bq.w;
  }
#pragma unroll
  for (int p = 0; p < 8; ++p) {
#pragma unroll
    for (int kq = 0; kq < 4; ++kq) {
      v4f q = pw[p * 4 + kq];
      v2f w0; w0.x = q.x; w0.y = q.y;
      v2f w1; w1.x = q.z; w1.y = q.w;
      v2f a0; a0.x = in[2 * kq];     a0.y = in[2 * kq];
      v2f a1; a1.x = in[2 * kq + 1]; a1.y = in[2 * kq + 1];
      g[p] = __builtin_elementwise_fma(a0, w0, g[p]);
      g[p] = __builtin_elementwise_fma(a1, w1, g[p]);
    }
  }
#pragma unroll
  for (int q = 0; q < 2; ++q) {
    v2f iv = sig2(g[0 + q]);
    v2f fv = sig2(g[2 + q]);
    v2f gv = tanh2(g[4 + q]);
    v2f ov = sig2(g[6 + q]);
    c[q] = __builtin_elementwise_fma(fv, c[q], iv * gv);
    h[q] = ov * tanh2(c[q]);
  }
}

// ---------- repack Wout [28][800] into WMMA-B order ----------
__global__ void repack_wout_kernel(const float* __restrict__ Wout,
                                   v2f* __restrict__ WP) {
  int idx = blockIdx.x * blockDim.x + threadIdx.x;
  if (idx >= WP_ENTRIES) return;
  int c = idx & 31;
  int kp = (idx >> 5) & 1;
  int t = idx >> 6;
  v2f v; v.x = 0.f; v.y = 0.f;
  if (c < OUTN) {
    const float* p = Wout + c * 800 + 4 * t + 2 * kp;
    v.x = p[0];
    v.y = p[1];
  }
  WP[idx] = v;
}

// ---------- fused 2-layer LSTM + WMMA output GEMM ----------
__global__ void __launch_bounds__(128)
lstm_fused_kernel(const float* __restrict__ x,
                  const float* __restrict__ Wih0, const float* __restrict__ Whh0,
                  const float* __restrict__ bih0, const float* __restrict__ bhh0,
                  const float* __restrict__ Wih1, const float* __restrict__ Whh1,
                  const float* __restrict__ bih1, const float* __restrict__ bhh1,
                  const v2f* __restrict__ WP, const float* __restrict__ bout,
                  float* __restrict__ out) {
  // LDS layout (floats): [0,128) layer0 packed pw | [128,256) layer1 packed pw
  //                      [256,272) bias0 | [272,288) bias1
  __shared__ __align__(16) float sw[288];
  const int tid = threadIdx.x;

  for (int i = tid; i < 288; i += blockDim.x) {
    float v;
    if (i < 256) {
      int layer = i >> 7;
      int f = i & 127;
      int e = f >> 1, comp = f & 1;
      int p = e >> 3, k = e & 7;
      int j = 2 * p + comp;
      const float* Wih = layer ? Wih1 : Wih0;
      const float* Whh = layer ? Whh1 : Whh0;
      v = (k < 4) ? Wih[j * 4 + k] : Whh[j * 4 + (k - 4)];
    } else if (i < 272) {
      v = bih0[i - 256] + bhh0[i - 256];
    } else {
      v = bih1[i - 272] + bhh1[i - 272];
    }
    sw[i] = v;
  }
  __syncthreads();

  const v4f* sPW0 = (const v4f*)(sw);         // streamed from LDS each step
  const v4f* sPW1 = (const v4f*)(sw + 128);   // copied to registers below
  const v4f* sPB0 = (const v4f*)(sw + 256);
  const v4f* sPB1 = (const v4f*)(sw + 272);

  // Layer-1 weights resident in registers (128 VGPRs)
  v4f pw1r[32];
#pragma unroll
  for (int q = 0; q < 32; ++q) pw1r[q] = sPW1[q];

  const int lane = tid & 31;
  const int col = lane & 15;
  const bool hiHalf = (lane & 16) != 0;
  const int waveBase = blockIdx.x * blockDim.x + (tid & ~31);
  const int b = waveBase + lane;

  const v4f* __restrict__ xp = (const v4f*)(x + (size_t)b * T_STEPS * HID);

  v2f h0[2], c0[2], h1[2], c1[2];
#pragma unroll
  for (int q = 0; q < 2; ++q) {
    h0[q] = (v2f)0.f; c0[q] = (v2f)0.f;
    h1[q] = (v2f)0.f; c1[q] = (v2f)0.f;
  }

  // WMMA accumulators: out[32 batch][32 col] as 4 C tiles, init with bias
  float bo0 = bout[col];
  float bo1 = (col + 16 < OUTN) ? bout[col + 16] : 0.f;
  v8f acc00, acc01, acc10, acc11;
#pragma unroll
  for (int r = 0; r < 8; ++r) {
    acc00[r] = bo0; acc10[r] = bo0;
    acc01[r] = bo1; acc11[r] = bo1;
  }

  const int kp = hiHalf ? 1 : 0;
  const v2f* __restrict__ wpLane = WP + kp * 32 + col;

  for (int t = 0; t < T_STEPS; ++t) {
    v4f xv = xp[t];
    __builtin_prefetch((const void*)(xp + t + 8), 0, 0);

    float in0[8] = {xv.x, xv.y, xv.z, xv.w,
                    h0[0].x, h0[0].y, h0[1].x, h0[1].y};
    cell_pk(sPW0, sPB0, in0, h0, c0);            // layer 0 (weights via LDS)

    float in1[8] = {h0[0].x, h0[0].y, h0[1].x, h0[1].y,
                    h1[0].x, h1[0].y, h1[1].x, h1[1].y};
    cell_pk(pw1r, sPB1, in1, h1, c1);            // layer 1 (weights in VGPRs)

    // A tiles (16x4 f32): lane=M, lane-half splits K{0,1}|{2,3}.
    // Only half of each tile is remote -> 4 half-swaps via ds_swizzle SWAPX16.
    float y0 = h1[0].x, y1 = h1[0].y, y2 = h1[1].x, y3 = h1[1].y;
    float s0 = swap16(y0), s1 = swap16(y1), s2 = swap16(y2), s3 = swap16(y3);
    v2f A0, A1;
    A0.x = hiHalf ? s2 : y0;  A0.y = hiHalf ? s3 : y1;
    A1.x = hiHalf ? y2 : s0;  A1.y = hiHalf ? y3 : s1;

    // B tiles from repacked Wout (coalesced, L2-resident)
    const v2f* wr = wpLane + t * 64;
    v2f B0 = wr[0];
    v2f B1 = wr[16];

    acc00 = __builtin_amdgcn_wmma_f32_16x16x4_f32(false, A0, false, B0, (short)0, acc00, false, false);
    acc01 = __builtin_amdgcn_wmma_f32_16x16x4_f32(false, A0, false, B1, (short)0, acc01, false, false);
    acc10 = __builtin_amdgcn_wmma_f32_16x16x4_f32(false, A1, false, B0, (short)0, acc10, false, false);
    acc11 = __builtin_amdgcn_wmma_f32_16x16x4_f32(false, A1, false, B1, (short)0, acc11, false, false);

    // Block LICM of the LDS weight loads (keeps VGPRs < 256, no MSB juggling),
    // placed at loop end so in-iteration scheduling stays free.
    asm volatile("" ::: "memory");
  }

  // Store C tiles: VGPR r -> M=r (+8 hi half); lane -> N=col (+16 for tile n1)
  float* outp0 = out + (size_t)(waveBase + (hiHalf ? 8 : 0)) * OUTN + col;
#pragma unroll
  for (int r = 0; r < 8; ++r) {
    outp0[(size_t)r * OUTN] = acc00[r];
    outp0[(size_t)(r + 16) * OUTN] = acc10[r];
    if (col < OUTN - 16) {
      outp0[(size_t)r * OUTN + 16] = acc01[r];
      outp0[(size_t)(r + 16) * OUTN + 16] = acc11[r];
    }
  }
}

extern "C" void kernel_launch(void* const* d_in, const int* in_sizes, int n_in,
                              void* d_out, int out_size, void* d_ws, size_t ws_size,
                              hipStream_t stream) {
  const float* x    = (const float*)d_in[0];
  const float* Wih0 = (const float*)d_in[1];
  const float* Whh0 = (const float*)d_in[2];
  const float* bih0 = (const float*)d_in[3];
  const float* bhh0 = (const float*)d_in[4];
  const float* Wih1 = (const float*)d_in[5];
  const float* Whh1 = (const float*)d_in[6];
  const float* bih1 = (const float*)d_in[7];
  const float* bhh1 = (const float*)d_in[8];
  const float* Wout = (const float*)d_in[9];
  const float* bout = (const float*)d_in[10];
  float* out = (float*)d_out;
  v2f* WP = (v2f*)d_ws; // 102400 bytes

  int B = in_sizes[0] / (T_STEPS * HID); // 32768

  repack_wout_kernel<<<(WP_ENTRIES + 255) / 256, 256, 0, stream>>>(Wout, WP);
  lstm_fused_kernel<<<B / 128, 128, 0, stream>>>(x, Wih0, Whh0, bih0, bhh0,
                                                 Wih1, Whh1, bih1, bhh1,
                                                 WP, bout, out);
}